// EijMatrix_test_87909390615221
// MI455X (gfx1250) — compile-verified
//
#include <hip/hip_runtime.h>

#ifndef __has_builtin
#define __has_builtin(x) 0
#endif

#define N_ATOMS 100
#define NC2     4950          // N*(N-1)/2
#define NCOORD  (N_ATOMS * 3) // 300 floats per batch row
#define THREADS 256

// ---------------------------------------------------------------------------
// CDNA5 async global->LDS copy (tracked by ASYNCcnt).
// Builtin signature (from hipcc diagnostic): first param is
// 'int __vector(2) addrspace(1)*', second 'int __vector(2) addrspace(3)*'.
// Rows are 8-byte aligned (19800B / 1200B strides), so B64 is the widest
// universally-aligned async width.
// ---------------------------------------------------------------------------
typedef int v2i __attribute__((ext_vector_type(2)));
typedef __attribute__((address_space(1))) v2i* gptr_v2i;
typedef __attribute__((address_space(3))) v2i* lptr_v2i;

__device__ __forceinline__ void async_copy_b64(const float* g, float* l) {
#if __has_builtin(__builtin_amdgcn_global_load_async_to_lds_b64)
  __builtin_amdgcn_global_load_async_to_lds_b64(
      (gptr_v2i)g, (lptr_v2i)l, /*imm offset=*/0, /*cpol=*/0);
#else
  // Synchronous fallback (still correct, just not async).
  *(float2*)l = *(const float2*)g;
#endif
}

__device__ __forceinline__ void wait_async_zero() {
#if __has_builtin(__builtin_amdgcn_s_wait_asynccnt)
  __builtin_amdgcn_s_wait_asynccnt(0);
#else
  asm volatile("s_wait_asynccnt 0" ::: "memory");
#endif
}

// ---------------------------------------------------------------------------
// One workgroup per batch row b:
//   out[b] = sum over strict-lower-tri pairs p=(i,j) of flat[b,p] / r_ij
// ---------------------------------------------------------------------------
__global__ __launch_bounds__(THREADS)
void eij_energy_kernel(const float* __restrict__ coords,      // [B, N, 3]
                       const float* __restrict__ flat,        // [B, NC2]
                       float* __restrict__ out) {             // [B]
  __shared__ __align__(16) float s_flat[NC2];
  __shared__ __align__(16) float s_coords[NCOORD];
  __shared__ float s_red[THREADS / 32];

  const int b   = blockIdx.x;
  const int tid = threadIdx.x;

  // ---- Async-stage the entire flat row (2475 x b64) and coords (150 x b64)
  const float* gflat = flat   + (size_t)b * NC2;
  const float* gcrd  = coords + (size_t)b * NCOORD;

  for (int u = tid; u < NC2 / 2; u += THREADS)
    async_copy_b64(gflat + 2 * u, &s_flat[2 * u]);
  for (int u = tid; u < NCOORD / 2; u += THREADS)
    async_copy_b64(gcrd + 2 * u, &s_coords[2 * u]);

  wait_async_zero();   // this wave's async loads landed in LDS
  __syncthreads();     // ... and every other wave's too

  // ---- Compute: each thread handles ~19 pairs
  float acc = 0.0f;
  for (int p = tid; p < NC2; p += THREADS) {
    // Invert row-major strict-lower-triangle index: T(i)=i*(i-1)/2 <= p < T(i+1)
    int i = (int)((1.0f + sqrtf(8.0f * (float)p + 1.0f)) * 0.5f);
    while ((i * (i - 1)) / 2 > p) --i;          // float-rounding fixups
    while ((i * (i + 1)) / 2 <= p) ++i;
    const int j = p - (i * (i - 1)) / 2;

    const float dx = s_coords[3 * i + 0] - s_coords[3 * j + 0];
    const float dy = s_coords[3 * i + 1] - s_coords[3 * j + 1];
    const float dz = s_coords[3 * i + 2] - s_coords[3 * j + 2];
    const float d2 = dx * dx + dy * dy + dz * dz;

    acc = fmaf(s_flat[p], 1.0f / sqrtf(d2), acc);  // IEEE-accurate 1/r
  }

  // ---- Reduction: wave32 shuffle tree, then cross-wave via LDS
  for (int off = 16; off > 0; off >>= 1)
    acc += __shfl_down(acc, off, 32);

  const int wid  = tid >> 5;
  const int lane = tid & 31;
  if (lane == 0) s_red[wid] = acc;
  __syncthreads();

  if (tid == 0) {
    float s = 0.0f;
#pragma unroll
    for (int w = 0; w < THREADS / 32; ++w) s += s_red[w];
    out[b] = s;
  }
}

// ---------------------------------------------------------------------------
extern "C" void kernel_launch(void* const* d_in, const int* in_sizes, int n_in,
                              void* d_out, int out_size, void* d_ws, size_t ws_size,
                              hipStream_t stream) {
  const float* coords = (const float*)d_in[0];  // [B, N, 3] fp32
  const float* flat   = (const float*)d_in[1];  // [B, NC2]  fp32
  float*       out    = (float*)d_out;          // [B, 1]    fp32

  const int B = in_sizes[1] / NC2;              // 2048
  eij_energy_kernel<<<dim3(B), dim3(THREADS), 0, stream>>>(coords, flat, out);
}